// CrossAttentionLayer_69200513073442
// MI455X (gfx1250) — compile-verified
//
#include <hip/hip_runtime.h>

typedef __attribute__((ext_vector_type(2))) float v2f;
typedef __attribute__((ext_vector_type(8))) float v8f;

static constexpr int kB = 4;
static constexpr int kN = 2048;
static constexpr int kE = 1024;
static constexpr int kH = 16;
static constexpr int kD = 64;
static constexpr float kScale = 0.125f;              // D^-0.5
static constexpr float kEps = 1.1920929e-07f;        // float32 eps

static __device__ __forceinline__ v8f wmma4(v2f a, v2f b, v8f c) {
  // V_WMMA_F32_16X16X4_F32: D = A(16x4) * B(4x16) + C(16x16)
  return __builtin_amdgcn_wmma_f32_16x16x4_f32(
      /*neg_a=*/false, a, /*neg_b=*/false, b,
      /*c_mod=*/(short)0, c, /*reuse_a=*/false, /*reuse_b=*/false);
}

// --- CDNA5 async global->LDS staging (ASYNCcnt path), with safe fallback ----
#if __has_builtin(__builtin_amdgcn_global_load_async_to_lds_b128) && \
    __has_builtin(__builtin_amdgcn_s_wait_asynccnt)
#define USE_ASYNC_LDS 1
#else
#define USE_ASYNC_LDS 0
#endif

#if USE_ASYNC_LDS
typedef int vi4 __attribute__((vector_size(4 * sizeof(int))));
typedef __attribute__((address_space(1))) vi4* as1_vi4p;   // global
typedef __attribute__((address_space(3))) vi4* as3_vi4p;   // LDS
#endif

static __device__ __forceinline__ void copy16(const float* __restrict__ g,
                                              float* __restrict__ l) {
#if USE_ASYNC_LDS
  __builtin_amdgcn_global_load_async_to_lds_b128(
      (as1_vi4p)g, (as3_vi4p)l, /*offset=*/0, /*cpol=*/0);
#else
  float4 v = *reinterpret_cast<const float4*>(g);
  l[0] = v.x; l[1] = v.y; l[2] = v.z; l[3] = v.w;
#endif
}

static __device__ __forceinline__ void async_join() {
#if USE_ASYNC_LDS
  __builtin_amdgcn_s_wait_asynccnt(0);
#endif
}

// ---------------------------------------------------------------------------
// Generic GEMM: C[M x Nc] = A[M x K] @ W[K x Nc] + bias[Nc]
// Block = 128 threads (4 waves), block tile 64x128, each wave 32x64 (2x4 WMMA
// tiles): per k-step 6 LDS frag loads feed 8 WMMAs.
// Requires M % 64 == 0, Nc % 128 == 0, K % 32 == 0.
// ---------------------------------------------------------------------------
__global__ __launch_bounds__(128)
void gemm_bias_kernel(const float* __restrict__ A, const float* __restrict__ W,
                      const float* __restrict__ bias, float* __restrict__ C,
                      int M, int K, int Nc) {
  __shared__ float As[64][36];    // +4 pad: 16B-aligned rows, conflict-free frags
  __shared__ float Ws[32][132];

  const int tid  = threadIdx.x;
  const int wave = tid >> 5;
  const int lane = tid & 31;
  const int lm   = lane & 15;
  const int kh   = (lane >> 4) << 1;   // K offset 0/2 (A/B frag layout)
  const int gh   = (lane >> 4) << 3;   // M offset 0/8 (C/D frag layout)
  const int wm   = (wave >> 1) * 32;
  const int wn   = (wave & 1) * 64;
  const int m0   = blockIdx.y * 64;
  const int n0   = blockIdx.x * 128;

  v8f acc[2][4] = {{{}, {}, {}, {}}, {{}, {}, {}, {}}};

  for (int kc = 0; kc < K; kc += 32) {
    // Stage A tile 64x32 (512 x 16B, 4 per thread)
#pragma unroll
    for (int i = 0; i < 4; i++) {
      int fid = tid + i * 128;
      int r = fid >> 3;
      int c4 = (fid & 7) << 2;
      copy16(&A[(size_t)(m0 + r) * K + kc + c4], &As[r][c4]);
    }
    // Stage W tile 32x128 (1024 x 16B, 8 per thread)
#pragma unroll
    for (int i = 0; i < 8; i++) {
      int fid = tid + i * 128;
      int r = fid >> 5;
      int c4 = (fid & 31) << 2;
      copy16(&W[(size_t)(kc + r) * Nc + n0 + c4], &Ws[r][c4]);
    }
    async_join();
    __syncthreads();

#pragma unroll
    for (int kk = 0; kk < 32; kk += 4) {
      v2f a0, a1;
      a0[0] = As[wm + lm][kk + kh];      a0[1] = As[wm + lm][kk + kh + 1];
      a1[0] = As[wm + 16 + lm][kk + kh]; a1[1] = As[wm + 16 + lm][kk + kh + 1];
#pragma unroll
      for (int j = 0; j < 4; j++) {
        v2f b;
        b[0] = Ws[kk + kh][wn + j * 16 + lm];
        b[1] = Ws[kk + kh + 1][wn + j * 16 + lm];
        acc[0][j] = wmma4(a0, b, acc[0][j]);
        acc[1][j] = wmma4(a1, b, acc[1][j]);
      }
    }
    __syncthreads();
  }

#pragma unroll
  for (int i = 0; i < 2; i++)
#pragma unroll
    for (int j = 0; j < 4; j++) {
      int colg = n0 + wn + j * 16 + lm;
      float bb = bias[colg];
#pragma unroll
      for (int g = 0; g < 8; g++) {
        int row = m0 + wm + i * 16 + g + gh;
        C[(size_t)row * Nc + colg] = acc[i][j][g] + bb;
      }
    }
}

// ---------------------------------------------------------------------------
// QKV interleaved rearrange + RMSNorm. One wave per (b,n,h); 2 dims per lane.
// c = h*192 + 3*d + j indexes concat(q_proj, kv_proj).
// ---------------------------------------------------------------------------
__global__ __launch_bounds__(256)
void qkv_rearrange_kernel(const float* __restrict__ P1, const float* __restrict__ P2,
                          const float* __restrict__ qnw, const float* __restrict__ knw,
                          float* __restrict__ Q, float* __restrict__ Kd,
                          float* __restrict__ V) {
  const int wave = threadIdx.x >> 5;
  const int lane = threadIdx.x & 31;
  const long idx = (long)blockIdx.x * 8 + wave;   // 0 .. B*N*H-1
  const int h = (int)(idx % kH);
  const long bn = idx / kH;

  const int base = h * 192 + 6 * lane;            // 3*d0 with d0 = 2*lane
  auto fetch = [&](int c) -> float {
    return (c < kE) ? P1[bn * kE + c] : P2[bn * (2 * kE) + (c - kE)];
  };
  float q0 = fetch(base + 0) * kScale;
  float k0 = fetch(base + 1);
  float v0 = fetch(base + 2);
  float q1 = fetch(base + 3) * kScale;
  float k1 = fetch(base + 4);
  float v1 = fetch(base + 5);

  float sq = q0 * q0 + q1 * q1;
  float sk = k0 * k0 + k1 * k1;
#pragma unroll
  for (int off = 16; off > 0; off >>= 1) {
    sq += __shfl_xor(sq, off, 32);
    sk += __shfl_xor(sk, off, 32);
  }
  float rq = rsqrtf(sq * (1.0f / (float)kD) + kEps);
  float rk = rsqrtf(sk * (1.0f / (float)kD) + kEps);

  const int d0 = lane * 2, d1 = d0 + 1;
  const long b = bn / kN, n = bn % kN;
  const long obase = ((b * kH + h) * (long)kN + n) * kD;
  Q[obase + d0] = q0 * rq * qnw[d0];
  Q[obase + d1] = q1 * rq * qnw[d1];
  Kd[obase + d0] = k0 * rk * knw[d0];
  Kd[obase + d1] = k1 * rk * knw[d1];
  V[obase + d0] = v0;
  V[obase + d1] = v1;
}

// ---------------------------------------------------------------------------
// Flash attention: block = 4 waves, each wave owns a 16-query tile.
// 16-key K/V tiles async-staged in LDS; online softmax in C-matrix lane layout.
// ---------------------------------------------------------------------------
__global__ __launch_bounds__(128)
void flash_attn_kernel(const float* __restrict__ Q, const float* __restrict__ K,
                       const float* __restrict__ V, float* __restrict__ O) {
  __shared__ float Ks[16][68];
  __shared__ float Vs[16][68];
  __shared__ float Ps[4][16][20];   // per-wave P relayout (8B-aligned b64 reads)

  const int tid  = threadIdx.x;
  const int wave = tid >> 5;
  const int lane = tid & 31;
  const int lm   = lane & 15;
  const int kh   = (lane >> 4) << 1;
  const int gh   = (lane >> 4) << 3;
  const int bh   = blockIdx.y;            // b*H + h
  const int b    = bh >> 4, h = bh & 15;
  const int n0   = blockIdx.x * 64 + wave * 16;

  const float* Qb = Q + (size_t)bh * kN * kD;
  const float* Kb = K + (size_t)bh * kN * kD;
  const float* Vb = V + (size_t)bh * kN * kD;

  // Q tile 16x64 as 16 A-fragments (K-steps of 4), kept in registers.
  v2f qf[16];
#pragma unroll
  for (int s = 0; s < 16; s++) {
    const float* p = &Qb[(size_t)(n0 + lm) * kD + 4 * s + kh];
    qf[s][0] = p[0];
    qf[s][1] = p[1];
  }

  v8f acc[4] = {{}, {}, {}, {}};
  float run_m[8], run_l[8];
#pragma unroll
  for (int g = 0; g < 8; g++) { run_m[g] = -__builtin_inff(); run_l[g] = 0.0f; }

  for (int kb = 0; kb < kN; kb += 16) {
    // Stage K,V tiles (16x64 each): 256 x 16B per matrix, 2 per thread.
#pragma unroll
    for (int i = 0; i < 2; i++) {
      int fid = tid + i * 128;
      int r = fid >> 4;
      int c4 = (fid & 15) << 2;
      copy16(&Kb[(size_t)(kb + r) * kD + c4], &Ks[r][c4]);
      copy16(&Vb[(size_t)(kb + r) * kD + c4], &Vs[r][c4]);
    }
    async_join();
    __syncthreads();

    // S = Q @ K^T  (16x16 scores), B-frag: K^T[dim][key] = Ks[key][dim]
    v8f S = {};
#pragma unroll
    for (int s = 0; s < 16; s++) {
      v2f bk;
      bk[0] = Ks[lm][4 * s + kh];
      bk[1] = Ks[lm][4 * s + kh + 1];
      S = wmma4(qf[s], bk, S);
    }

    // Online softmax per query row (row = g + gh, 16 keys across lanes lm)
    float corr[8];
#pragma unroll
    for (int g = 0; g < 8; g++) {
      float v = S[g];
      float m = v;
      m = fmaxf(m, __shfl_xor(m, 1, 32));
      m = fmaxf(m, __shfl_xor(m, 2, 32));
      m = fmaxf(m, __shfl_xor(m, 4, 32));
      m = fmaxf(m, __shfl_xor(m, 8, 32));
      float nm = fmaxf(run_m[g], m);
      float p  = __expf(v - nm);
      float c  = __expf(run_m[g] - nm);
      float l  = p;
      l += __shfl_xor(l, 1, 32);
      l += __shfl_xor(l, 2, 32);
      l += __shfl_xor(l, 4, 32);
      l += __shfl_xor(l, 8, 32);
      run_l[g] = run_l[g] * c + l;
      run_m[g] = nm;
      corr[g]  = c;
      Ps[wave][g + gh][lm] = p;   // row-major P for A-frag reload
    }
#pragma unroll
    for (int t = 0; t < 4; t++)
#pragma unroll
      for (int g = 0; g < 8; g++) acc[t][g] *= corr[g];

    // O += P(16x16) @ V(16x64): 4 K-steps x 4 dim-tiles
#pragma unroll
    for (int ks = 0; ks < 4; ks++) {
      v2f ap;
      ap[0] = Ps[wave][lm][4 * ks + kh];
      ap[1] = Ps[wave][lm][4 * ks + kh + 1];
#pragma unroll
      for (int t = 0; t < 4; t++) {
        v2f bv;
        bv[0] = Vs[4 * ks + kh][16 * t + lm];
        bv[1] = Vs[4 * ks + kh + 1][16 * t + lm];
        acc[t] = wmma4(ap, bv, acc[t]);
      }
    }
    __syncthreads();
  }

  // Normalize and write ctx in [B,N,E] layout: col = h*64 + dim
#pragma unroll
  for (int g = 0; g < 8; g++) {
    float inv = 1.0f / run_l[g];
    int row = n0 + g + gh;
    size_t rb = ((size_t)b * kN + row) * kE + h * kD;
#pragma unroll
    for (int t = 0; t < 4; t++) O[rb + 16 * t + lm] = acc[t][g] * inv;
  }
}

// ---------------------------------------------------------------------------
extern "C" void kernel_launch(void* const* d_in, const int* in_sizes, int n_in,
                              void* d_out, int out_size, void* d_ws, size_t ws_size,
                              hipStream_t stream) {
  (void)in_sizes; (void)n_in; (void)out_size; (void)ws_size;
  const float* hs  = (const float*)d_in[0];
  const float* ehs = (const float*)d_in[1];
  const float* Wq  = (const float*)d_in[2];
  const float* bq  = (const float*)d_in[3];
  const float* Wkv = (const float*)d_in[4];
  const float* bkv = (const float*)d_in[5];
  const float* qnw = (const float*)d_in[6];
  const float* knw = (const float*)d_in[7];
  const float* Wo  = (const float*)d_in[8];
  const float* bo  = (const float*)d_in[9];
  float* out = (float*)d_out;

  const int M = kB * kN;                       // 8192
  float* P1 = (float*)d_ws;                    // [M, E]
  float* P2 = P1 + (size_t)M * kE;             // [M, 2E]
  float* Qb = P2 + (size_t)M * 2 * kE;         // [B*H, N, D]
  float* Kb = Qb + (size_t)M * kE;
  float* Vb = Kb + (size_t)M * kE;
  float* ctx = P1;                             // reuse P1 region after rearrange

  dim3 blk(128);
  gemm_bias_kernel<<<dim3(kE / 128, M / 64), blk, 0, stream>>>(hs, Wq, bq, P1, M, kE, kE);
  gemm_bias_kernel<<<dim3(2 * kE / 128, M / 64), blk, 0, stream>>>(ehs, Wkv, bkv, P2, M, kE, 2 * kE);
  qkv_rearrange_kernel<<<dim3(kB * kN * kH / 8), dim3(256), 0, stream>>>(P1, P2, qnw, knw, Qb, Kb, Vb);
  flash_attn_kernel<<<dim3(kN / 64, kB * kH), blk, 0, stream>>>(Qb, Kb, Vb, ctx);
  gemm_bias_kernel<<<dim3(kE / 128, M / 64), blk, 0, stream>>>(ctx, Wo, bo, out, M, kE, kE);
}